// RejectionSampler_81003083203228
// MI455X (gfx1250) — compile-verified
//
#include <hip/hip_runtime.h>
#include <stdint.h>

// Problem constants (match reference: BATCH=64, K=4, VOCAB=128000)
#define BATCH   64
#define KSPEC   4
#define ROWS    (BATCH * (KSPEC + 1))   // 320
#define VOCAB   128000
#define SEG     5                        // segments per row
#define SEGLEN  (VOCAB / SEG)            // 25600 floats = 100 KB per segment
#define TPB     256                      // 8 wave32s
#define TILE    (TPB * 4)                // 1024 floats per tile
#define ITERS   (SEGLEN / TILE)          // 25 tiles
#define DEPTH   4                        // async pipeline depth (ring buffers)

// ---- CDNA5 async global->LDS copy (tracked by ASYNCcnt) ----
__device__ __forceinline__ void async_load_b128(unsigned lds_off, const float* gaddr) {
    // GV mode: per-lane LDS byte address in one VGPR, 64-bit global address in a VGPR pair.
    asm volatile("global_load_async_to_lds_b128 %0, %1, off"
                 :: "v"(lds_off), "v"(gaddr)
                 : "memory");
}

// s_wait_asynccnt with a small-constant immediate; the switch folds away after
// full unroll (n is compile-time constant at every call site).
__device__ __forceinline__ void wait_asynccnt(int n) {
    switch (n) {
        case 0:  asm volatile("s_wait_asynccnt 0x0" ::: "memory"); break;
        case 1:  asm volatile("s_wait_asynccnt 0x1" ::: "memory"); break;
        case 2:  asm volatile("s_wait_asynccnt 0x2" ::: "memory"); break;
        default: asm volatile("s_wait_asynccnt 0x3" ::: "memory"); break;
    }
}

__device__ __forceinline__ void amax_combine(float& bv, int& bi, float v, int idx) {
    // jnp.argmax semantics: strictly greater wins; on tie the smaller index wins.
    if (v > bv || (v == bv && idx < bi)) { bv = v; bi = idx; }
}

// Kernel 1: per-(row, segment) streaming argmax, 4-deep async-LDS ring pipeline.
__global__ __launch_bounds__(TPB)
void rs_argmax_partial(const float* __restrict__ logits,
                       float* __restrict__ part_v,
                       int*   __restrict__ part_i) {
    const int seg = blockIdx.x;   // 0..SEG-1
    const int row = blockIdx.y;   // 0..ROWS-1
    const int tid = threadIdx.x;

    __shared__ float buf[DEPTH][TILE];    // 16 KB ring
    __shared__ float red_v[TPB / 32];
    __shared__ int   red_i[TPB / 32];

    const float* rowp = logits + (size_t)row * VOCAB;
    const int    col0 = seg * SEGLEN + tid * 4;   // this thread's first column

    // LDS byte offset of this thread's 16B slot in ring buffer 0; flat addresses
    // of __shared__ carry the LDS offset in the low 32 bits. Slot s is +s*sizeof(buf[0]).
    const unsigned lds0    = (unsigned)(uintptr_t)&buf[0][tid * 4];
    const unsigned ldsStep = (unsigned)sizeof(buf[0]);   // 4096 B

    // Prologue: DEPTH-1 tiles in flight.
    #pragma unroll
    for (int k = 0; k < DEPTH - 1; ++k)
        async_load_b128(lds0 + (unsigned)k * ldsStep, rowp + col0 + k * TILE);

    float bv = -__builtin_inff();
    int   bi = 0x7fffffff;

    #pragma unroll
    for (int i = 0; i < ITERS; ++i) {
        // Keep DEPTH tiles in flight, branchlessly (guard folds after unroll).
        if (i + DEPTH - 1 < ITERS)
            async_load_b128(lds0 + (unsigned)((i + DEPTH - 1) % DEPTH) * ldsStep,
                            rowp + col0 + (i + DEPTH - 1) * TILE);
        // Outstanding after tile i completes: min(DEPTH-1, ITERS-1-i).
        wait_asynccnt((ITERS - 1 - i < DEPTH - 1) ? (ITERS - 1 - i) : (DEPTH - 1));

        const float4 v = *(const float4*)&buf[i % DEPTH][tid * 4];
        const int c = col0 + i * TILE;
        amax_combine(bv, bi, v.x, c);
        amax_combine(bv, bi, v.y, c + 1);
        amax_combine(bv, bi, v.z, c + 2);
        amax_combine(bv, bi, v.w, c + 3);
    }

    // wave32 reduction
    #pragma unroll
    for (int off = 16; off > 0; off >>= 1) {
        float ov = __shfl_down(bv, off, 32);
        int   oi = __shfl_down(bi, off, 32);
        amax_combine(bv, bi, ov, oi);
    }
    if ((tid & 31) == 0) { red_v[tid >> 5] = bv; red_i[tid >> 5] = bi; }
    __syncthreads();
    if (tid == 0) {
        #pragma unroll
        for (int w = 1; w < TPB / 32; ++w) amax_combine(bv, bi, red_v[w], red_i[w]);
        part_v[row * SEG + seg] = bv;
        part_i[row * SEG + seg] = bi;
    }
}

// Kernel 2: reduce segment partials -> per-row argmax ids, then greedy
// rejection-sampling acceptance. Single block; tiny.
__global__ __launch_bounds__(ROWS)
void rs_finalize(const float* __restrict__ part_v,
                 const int*   __restrict__ part_i,
                 const int*   __restrict__ spec,     // [BATCH, KSPEC]
                 float*       __restrict__ out) {    // [BATCH, KSPEC+1]
    __shared__ int ids[ROWS];
    const int t = threadIdx.x;   // row id, 0..ROWS-1

    float bv = part_v[t * SEG];
    int   bi = part_i[t * SEG];
    #pragma unroll
    for (int s = 1; s < SEG; ++s) {
        const float v  = part_v[t * SEG + s];
        const int   ix = part_i[t * SEG + s];
        if (v > bv || (v == bv && ix < bi)) { bv = v; bi = ix; }
    }
    ids[t] = bi;
    __syncthreads();

    if (t < BATCH) {
        bool ok = true;
        int  num_emitted = 1;                      // correction/bonus token
        #pragma unroll
        for (int i = 0; i < KSPEC; ++i) {
            ok = ok && (spec[t * KSPEC + i] == ids[t * (KSPEC + 1) + i]);
            num_emitted += ok ? 1 : 0;
        }
        #pragma unroll
        for (int p = 0; p <= KSPEC; ++p) {
            const int tok = ids[t * (KSPEC + 1) + p];
            out[t * (KSPEC + 1) + p] = (p < num_emitted) ? (float)tok : -1.0f;
        }
    }
}

extern "C" void kernel_launch(void* const* d_in, const int* in_sizes, int n_in,
                              void* d_out, int out_size, void* d_ws, size_t ws_size,
                              hipStream_t stream) {
    (void)in_sizes; (void)n_in; (void)out_size; (void)ws_size;
    const float* logits = (const float*)d_in[0];      // [ROWS, VOCAB] fp32
    const int*   spec   = (const int*)d_in[1];        // [BATCH, KSPEC] int32

    float* part_v = (float*)d_ws;                                     // ROWS*SEG floats
    int*   part_i = (int*)((char*)d_ws + ROWS * SEG * sizeof(float)); // ROWS*SEG ints

    dim3 grid(SEG, ROWS);
    rs_argmax_partial<<<grid, TPB, 0, stream>>>(logits, part_v, part_i);
    rs_finalize<<<1, ROWS, 0, stream>>>(part_v, part_i, spec, (float*)d_out);
}